// eTofts_85607288144023
// MI455X (gfx1250) — compile-verified
//
#include <hip/hip_runtime.h>
#include <stdint.h>

// DCE-MRI eTofts forward model, MI455X (gfx1250, wave32).
// Memory-bound streaming (~240 MB @ 23.3 TB/s ~ 10us). WMMA is inapplicable
// (per-row IIR recurrence, no shared matrix operand); the CDNA5 feature that
// pays here is the async global->LDS copy engine (ASYNCcnt) used as a
// double-buffered prefetch pipe overlapping the HBM stream with VALU work.

#define TR_C      0.005f
#define DELTT_C   0.075f                 // 4.5/60
#define R1RELAX_C 4.5f
#define SINA_C    0.25881904510252074f   // sin(15 deg)
#define COSA_C    0.96592582628906831f   // cos(15 deg)

namespace {

constexpr int TT     = 112;  // timepoints
constexpr int CHUNK  = 28;   // floats per chunk (112/4), 112 bytes -> 7 x b128
constexpr int BLK    = 256;  // threads per block = 8 wave32

typedef int v4i __attribute__((vector_size(16)));
typedef __attribute__((address_space(1))) v4i* gptr_b128;
typedef __attribute__((address_space(3))) v4i* lptr_b128;

// ---- CDNA5 async global->LDS copy (16B per lane), ASYNCcnt-tracked ----
__device__ __forceinline__ void async_b128(const float* gp, float* lp) {
#if __has_builtin(__builtin_amdgcn_global_load_async_to_lds_b128)
  __builtin_amdgcn_global_load_async_to_lds_b128(
      (gptr_b128)(gp),
      (lptr_b128)(lp),
      /*imm offset*/ 0, /*cpol*/ 0);
#else
  uint32_t loff = (uint32_t)(uintptr_t)(__attribute__((address_space(3))) void*)lp;
  uint64_t gaddr = (uint64_t)(uintptr_t)gp;
  asm volatile("global_load_async_to_lds_b128 %0, %1, off"
               :: "v"(loff), "v"(gaddr) : "memory");
#endif
}

template <int N>
__device__ __forceinline__ void wait_asynccnt() {
#if __has_builtin(__builtin_amdgcn_s_wait_asynccnt)
  __builtin_amdgcn_s_wait_asynccnt(N);
#else
  asm volatile("s_wait_asynccnt %0" :: "i"(N) : "memory");
#endif
}

// WAR fence before re-staging into an LDS buffer we just read from:
// forces outstanding ds_load completion AND acts as a compiler barrier.
__device__ __forceinline__ void lds_read_fence() {
  asm volatile("s_wait_dscnt 0" ::: "memory");
}

__global__ __launch_bounds__(BLK) void etofts_kernel(
    const float* __restrict__ param,   // [B,3]
    const float* __restrict__ T10,     // [B,1]
    const float* __restrict__ cp,      // [B,112]
    float* __restrict__ out,           // [B,112]
    int B)
{
  // 2 x (256 rows x 28 floats) = 2 x 28KB double buffer; 16B aligned rows.
  __shared__ __align__(16) float lds[2][BLK * CHUNK];

  const int tid = threadIdx.x;
  const int row = blockIdx.x * BLK + tid;
  if (row >= B) return;

  const float* grow  = cp  + (size_t)row * TT;   // this thread's cp row
  float*       orow  = out + (size_t)row * TT;
  float*       lrow0 = &lds[0][tid * CHUNK];     // byte off tid*112 (16B aligned)
  float*       lrow1 = &lds[1][tid * CHUNK];

  // ---- stage: 7 async b128 ops copy one 112B row-chunk into LDS ----
  auto stage = [&](int ch, float* lrow) {
#pragma unroll
    for (int q = 0; q < CHUNK / 4; ++q)
      async_b128(grow + ch * CHUNK + q * 4, lrow + q * 4);
  };

  // ---- per-voxel parameters (matches reference scaling/clamping) ----
  const float ktr = fminf(fmaxf(param[row * 3 + 0] * 0.2f, 1e-5f),   0.2f);
  const float vp  = fminf(fmaxf(param[row * 3 + 1] * 0.1f, 0.0005f), 0.1f);
  const float ve  = fminf(fmaxf(param[row * 3 + 2] * 0.6f, 0.04f),   0.6f);
  const float invT10 = 1.0f / T10[row];
  const float decay  = __expf(-(ktr / ve) * DELTT_C);

  float c = 0.0f;  // recurrence carry

  auto sig = [&](float cpv) -> float {
    c = c * decay + cpv * DELTT_C;                 // IIR scan step
    const float ct = vp * cpv + ktr * c;           // ce = ktr*c
    const float R1 = invT10 + R1RELAX_C * ct;
    const float E  = __expf(-TR_C * R1);           // v_exp_f32
    const float dn = 1.0f - E * COSA_C;
    return (1.0f - E) * (SINA_C * 20.0f) * __builtin_amdgcn_rcpf(dn);
  };

  auto process = [&](int ch, const float* lrow) {
#pragma unroll
    for (int q = 0; q < CHUNK / 4; ++q) {
      const float4 v = *reinterpret_cast<const float4*>(lrow + q * 4);
      float4 o;
      o.x = sig(v.x);
      o.y = sig(v.y);
      o.z = sig(v.z);
      o.w = sig(v.w);
      *reinterpret_cast<float4*>(orow + ch * CHUNK + q * 4) = o;  // b128 store
    }
  };

  // ---- software pipeline: ASYNCcnt completes in issue order, so after
  // issuing chunks (k, k+1), waiting asynccnt<=7 guarantees chunk k landed ----
  stage(0, lrow0);
  stage(1, lrow1);

  wait_asynccnt<7>();  process(0, lrow0);
  lds_read_fence();    stage(2, lrow0);

  wait_asynccnt<7>();  process(1, lrow1);
  lds_read_fence();    stage(3, lrow1);

  wait_asynccnt<7>();  process(2, lrow0);
  wait_asynccnt<0>();  process(3, lrow1);
}

} // anonymous namespace

extern "C" void kernel_launch(void* const* d_in, const int* in_sizes, int n_in,
                              void* d_out, int out_size, void* d_ws, size_t ws_size,
                              hipStream_t stream) {
  (void)n_in; (void)d_ws; (void)ws_size; (void)out_size;
  const float* param = (const float*)d_in[0];  // [B,3]
  const float* T10   = (const float*)d_in[1];  // [B,1]
  const float* cp    = (const float*)d_in[2];  // [B,112]
  float* out = (float*)d_out;                  // [B,112]

  const int B = in_sizes[0] / 3;
  const int grid = (B + BLK - 1) / BLK;
  etofts_kernel<<<grid, BLK, 0, stream>>>(param, T10, cp, out, B);
}